// GraphAttentionLayer_41180146434555
// MI455X (gfx1250) — compile-verified
//
#include <hip/hip_runtime.h>
#include <math.h>

// ---------------------------------------------------------------------------
// GAT layer, fused flash-attention formulation for MI455X (gfx1250, wave32).
//   Wh  = h @ W                         (8192x256 @ 256x128)
//   e   = leaky_relu(Wh @ Wh^T, 0.2)    (8192x8192, never materialized)
//   att = softmax(e, axis=1)            (streaming/online softmax)
//   out = att @ Wh                      (8192x128)
// fp32 WMMA (V_WMMA_F32_16X16X4_F32) for all matmuls; KV tiles are streamed
// into LDS with double-buffered GLOBAL_LOAD_ASYNC_TO_LDS_B64 (ASYNCcnt).
// ---------------------------------------------------------------------------

typedef float v2f __attribute__((ext_vector_type(2)));
typedef float v8f __attribute__((ext_vector_type(8)));

#define N_ROWS 8192
#define IN_F   256
#define OUT_F  128
#define NEG_SLOPE 0.2f

#define BM_WAVES 4          // waves per workgroup in the attention kernel
#define KV_PAD   130        // 128 + 2 : bank-conflict-free strided column reads
#define P_PAD    17         // 16 + 1  : bank-conflict-free transposed reads

static __device__ __forceinline__ v8f wmma_f32(v2f a, v2f b, v8f c) {
  // D = A(16x4,f32) * B(4x16,f32) + C(16x16,f32)
  return __builtin_amdgcn_wmma_f32_16x16x4_f32(
      /*neg_a=*/false, a, /*neg_b=*/false, b,
      /*c_mod=*/(short)0, c, /*reuse_a=*/false, /*reuse_b=*/false);
}

static __device__ __forceinline__ void wait_async0() {
#if defined(__has_builtin)
#if __has_builtin(__builtin_amdgcn_s_wait_asynccnt)
  __builtin_amdgcn_s_wait_asynccnt(0);
#else
  asm volatile("s_wait_asynccnt 0" ::: "memory");
#endif
#else
  asm volatile("s_wait_asynccnt 0" ::: "memory");
#endif
}

// One async direct-to-LDS 8-byte copy.  GVS mode: 64-bit SGPR base +
// 32-bit VGPR byte offset; the immediate offset is added to BOTH the
// global address and the LDS address (per CDNA5 ISA).
#define ASYNC_LD8(ldsoff, goff, base, OFF)                                   \
  asm volatile("global_load_async_to_lds_b64 %0, %1, %2 offset:" OFF        \
               :: "v"(ldsoff), "v"(goff), "s"(base) : "memory")

// ---------------------------------------------------------------------------
// Kernel 1: Wh = h @ W.  One wave per 16x16 output tile; dual accumulator
// chains break the WMMA RAW dependency.  Grid: 512 blocks x 256 threads.
// ---------------------------------------------------------------------------
__global__ __launch_bounds__(256) void gat_gemm_wh(
    const float* __restrict__ h, const float* __restrict__ W,
    float* __restrict__ Wh) {
  const int lane = threadIdx.x & 31;
  const int wave = threadIdx.x >> 5;   // 0..7 -> column tile
  const int lo   = lane & 15;
  const int hi   = lane >> 4;

  const int mt = blockIdx.x;           // 0..511 row tile
  const int m  = mt * 16 + lo;         // A-fragment row for this lane
  const int n  = wave * 16 + lo;       // B/C-fragment column for this lane

  v8f acc0 = {}, acc1 = {};
#pragma unroll 4
  for (int k = 0; k < IN_F; k += 8) {
    const int ka = k + 2 * hi;
    v2f a0, b0, a1, b1;
    a0.x = h[m * IN_F + ka];
    a0.y = h[m * IN_F + ka + 1];
    b0.x = W[ka * OUT_F + n];
    b0.y = W[(ka + 1) * OUT_F + n];
    a1.x = h[m * IN_F + ka + 4];
    a1.y = h[m * IN_F + ka + 5];
    b1.x = W[(ka + 4) * OUT_F + n];
    b1.y = W[(ka + 5) * OUT_F + n];
    acc0 = wmma_f32(a0, b0, acc0);
    acc1 = wmma_f32(a1, b1, acc1);
  }

#pragma unroll
  for (int r = 0; r < 8; ++r) {
    Wh[(mt * 16 + r + 8 * hi) * OUT_F + n] = acc0[r] + acc1[r];
  }
}

// ---------------------------------------------------------------------------
// Kernel 2: fused attention.  Each wave owns a 16-row query tile (Q cached in
// registers as A-fragments); the workgroup streams 16-row K/V tiles through a
// double-buffered LDS pipeline fed by async direct-to-LDS loads.
// Grid: 128 blocks x 128 threads (4 waves -> 64 query rows per block).
// ---------------------------------------------------------------------------
__global__ __launch_bounds__(128) void gat_attention(
    const float* __restrict__ Wh, float* __restrict__ out) {
  __shared__ alignas(16) float kv[2][16 * KV_PAD];  // double-buffered KV tile
  __shared__ float pbuf[BM_WAVES][16 * P_PAD];      // per-wave prob tile

  const int lane = threadIdx.x & 31;
  const int wave = threadIdx.x >> 5;
  const int lo   = lane & 15;
  const int hi   = lane >> 4;

  const int qrow_base = blockIdx.x * (BM_WAVES * 16) + wave * 16;
  const int qrow      = qrow_base + lo;

  // Async-load assignment: 128 threads cover a 16x128 f32 tile; thread
  // handles row (tid>>3), a 16-float (64 B) strip starting at col (tid&7)*16,
  // as 8 x b64 copies with immediate offsets 0..56.
  const int ldr = threadIdx.x >> 3;
  const int ldc = (threadIdx.x & 7) * 16;
  const unsigned lds_b0 =
      (unsigned)(unsigned long long)&kv[0][ldr * KV_PAD + ldc];
  const unsigned lds_b1 =
      (unsigned)(unsigned long long)&kv[1][ldr * KV_PAD + ldc];

  // Preload Q tile (16x128) as 32 A-fragments: qa[c] = Q[lo][4c + 2hi + {0,1}]
  v2f qa[32];
#pragma unroll
  for (int c = 0; c < 32; ++c) {
    const int f = 4 * c + 2 * hi;
    qa[c].x = Wh[qrow * OUT_F + f];
    qa[c].y = Wh[qrow * OUT_F + f + 1];
  }

  v8f o[8];
  float mrow[8], lrow[8];
#pragma unroll
  for (int t = 0; t < 8; ++t) o[t] = (v8f)0.f;
#pragma unroll
  for (int r = 0; r < 8; ++r) { mrow[r] = -INFINITY; lrow[r] = 0.f; }

  // Prologue: async-load tile 0 into buffer 0.
  {
    const unsigned g0 = (unsigned)((ldr * OUT_F + ldc) * 4);
    ASYNC_LD8(lds_b0, g0, Wh, "0");
    ASYNC_LD8(lds_b0, g0, Wh, "8");
    ASYNC_LD8(lds_b0, g0, Wh, "16");
    ASYNC_LD8(lds_b0, g0, Wh, "24");
    ASYNC_LD8(lds_b0, g0, Wh, "32");
    ASYNC_LD8(lds_b0, g0, Wh, "40");
    ASYNC_LD8(lds_b0, g0, Wh, "48");
    ASYNC_LD8(lds_b0, g0, Wh, "56");
  }

  int buf = 0;
  for (int j = 0; j < N_ROWS; j += 16, buf ^= 1) {
    wait_async0();     // this wave's async writes into kv[buf] landed
    __syncthreads();   // all waves' writes landed; kv[buf^1] no longer read

    // Prefetch next tile into the other buffer while we compute.
    if (j + 16 < N_ROWS) {
      const unsigned gn = (unsigned)(((j + 16 + ldr) * OUT_F + ldc) * 4);
      const unsigned ln = buf ? lds_b0 : lds_b1;
      ASYNC_LD8(ln, gn, Wh, "0");
      ASYNC_LD8(ln, gn, Wh, "8");
      ASYNC_LD8(ln, gn, Wh, "16");
      ASYNC_LD8(ln, gn, Wh, "24");
      ASYNC_LD8(ln, gn, Wh, "32");
      ASYNC_LD8(ln, gn, Wh, "40");
      ASYNC_LD8(ln, gn, Wh, "48");
      ASYNC_LD8(ln, gn, Wh, "56");
    }
    const float* __restrict__ kvb = &kv[buf][0];

    // S = Q @ K^T : two interleaved accumulator chains (break WMMA RAW).
    v8f s0 = {}, s1 = {};
#pragma unroll
    for (int c = 0; c < 32; c += 2) {
      const int f0 = 4 * c + 2 * hi;
      v2f b0, b1;
      b0.x = kvb[lo * KV_PAD + f0];
      b0.y = kvb[lo * KV_PAD + f0 + 1];
      b1.x = kvb[lo * KV_PAD + f0 + 4];
      b1.y = kvb[lo * KV_PAD + f0 + 5];
      s0 = wmma_f32(qa[c], b0, s0);
      s1 = wmma_f32(qa[c + 1], b1, s1);
    }

    // LeakyReLU + online softmax.  Register r on this lane = row (r + 8*hi),
    // column lo; row reductions = shfl_xor within each 16-lane half.
    float pr[8], scale[8];
#pragma unroll
    for (int r = 0; r < 8; ++r) {
      const float x = s0[r] + s1[r];
      pr[r] = (x >= 0.f) ? x : NEG_SLOPE * x;
    }
#pragma unroll
    for (int r = 0; r < 8; ++r) {
      float v = pr[r];
      v = fmaxf(v, __shfl_xor(v, 8, 32));
      v = fmaxf(v, __shfl_xor(v, 4, 32));
      v = fmaxf(v, __shfl_xor(v, 2, 32));
      v = fmaxf(v, __shfl_xor(v, 1, 32));
      const float newm = fmaxf(mrow[r], v);
      scale[r] = __expf(mrow[r] - newm);
      mrow[r]  = newm;
    }
#pragma unroll
    for (int r = 0; r < 8; ++r) {
      float p = __expf(pr[r] - mrow[r]);
      pr[r] = p;
      float v = p;
      v += __shfl_xor(v, 8, 32);
      v += __shfl_xor(v, 4, 32);
      v += __shfl_xor(v, 2, 32);
      v += __shfl_xor(v, 1, 32);
      lrow[r] = lrow[r] * scale[r] + v;
#pragma unroll
      for (int t = 0; t < 8; ++t) o[t][r] *= scale[r];
    }

    // Transpose P (C-layout -> A-layout) through per-wave LDS buffer.
#pragma unroll
    for (int r = 0; r < 8; ++r) {
      pbuf[wave][(r + 8 * hi) * P_PAD + lo] = pr[r];
    }

    // O += P @ V : A = P[m=lo][k], B = KV[k][f = 16t + lo], K = 16 keys.
    // 8 independent o[t] chains keep the matrix pipe busy.
#pragma unroll
    for (int c = 0; c < 4; ++c) {
      const int kk = 4 * c + 2 * hi;
      v2f a;
      a.x = pbuf[wave][lo * P_PAD + kk];
      a.y = pbuf[wave][lo * P_PAD + kk + 1];
#pragma unroll
      for (int t = 0; t < 8; ++t) {
        v2f b;
        b.x = kvb[kk * KV_PAD + t * 16 + lo];
        b.y = kvb[(kk + 1) * KV_PAD + t * 16 + lo];
        o[t] = wmma_f32(a, b, o[t]);
      }
    }
  }

  // Normalize by row sums and write out (row = qrow_base + r + 8*hi).
#pragma unroll
  for (int t = 0; t < 8; ++t) {
#pragma unroll
    for (int r = 0; r < 8; ++r) {
      out[(qrow_base + r + 8 * hi) * OUT_F + t * 16 + lo] = o[t][r] / lrow[r];
    }
  }
}

// ---------------------------------------------------------------------------
// Launch: inputs are (h, adj, W); adj is unused by the math.
// Workspace holds Wh (8192*128 fp32 = 4 MB).
// ---------------------------------------------------------------------------
extern "C" void kernel_launch(void* const* d_in, const int* in_sizes, int n_in,
                              void* d_out, int out_size, void* d_ws, size_t ws_size,
                              hipStream_t stream) {
  const float* h   = (const float*)d_in[0];
  const float* W   = (const float*)d_in[2];
  float*       out = (float*)d_out;
  float*       Wh  = (float*)d_ws;

  gat_gemm_wh<<<N_ROWS / 16, 256, 0, stream>>>(h, W, Wh);
  gat_attention<<<N_ROWS / (BM_WAVES * 16), BM_WAVES * 32, 0, stream>>>(Wh, out);
}